// PoseGraphOpt_75127567942091
// MI455X (gfx1250) — compile-verified
//
#include <hip/hip_runtime.h>
#include <hip/hip_bf16.h>
#include <stdint.h>

typedef uint32_t u32;
typedef uint64_t u64;
typedef __attribute__((ext_vector_type(4))) u32 u32x4;
typedef __attribute__((ext_vector_type(8))) int  i32x8;
typedef __attribute__((ext_vector_type(4))) int  i32x4;

#define BLOCK 256
#define POSE_DW (7 * BLOCK)   // 1792 dwords = 7168 B of poses staged per block

struct f3 { float x, y, z; };

__device__ __forceinline__ f3 cross3(f3 a, f3 b) {
    f3 r;
    r.x = a.y * b.z - a.z * b.y;
    r.y = a.z * b.x - a.x * b.z;
    r.z = a.x * b.y - a.y * b.x;
    return r;
}

struct quat { float x, y, z, w; };

__device__ __forceinline__ quat qmul(quat a, quat b) {
    quat r;
    r.x = a.w*b.x + a.x*b.w + a.y*b.z - a.z*b.y;
    r.y = a.w*b.y - a.x*b.z + a.y*b.w + a.z*b.x;
    r.z = a.w*b.z + a.x*b.y - a.y*b.x + a.z*b.w;
    r.w = a.w*b.w - a.x*b.x - a.y*b.y - a.z*b.z;
    return r;
}

__device__ __forceinline__ f3 qrot(quat q, f3 v) {
    f3 qv{q.x, q.y, q.z};
    f3 t = cross3(qv, v);
    t.x += t.x; t.y += t.y; t.z += t.z;          // t = 2*cross(qv, v)
    f3 c = cross3(qv, t);
    f3 r;
    r.x = v.x + q.w * t.x + c.x;
    r.y = v.y + q.w * t.y + c.y;
    r.z = v.z + q.w * t.z + c.z;
    return r;
}

__global__ __launch_bounds__(BLOCK)
void posegraph_err_kernel(const float* __restrict__ nodes,
                          const float* __restrict__ poses,
                          const long long* __restrict__ edges_pk,  // packed int32 pairs
                          float* __restrict__ out,
                          int n_edges, u32 total_pose_dw)
{
    __shared__ __align__(16) float sposes[POSE_DW];

    const int tid  = threadIdx.x;
    const int base = blockIdx.x * BLOCK;

    // ---- Stage 1: wave 0 kicks off the TDM DMA of this block's 7168 B pose
    //      slice into LDS. No wait yet -- let it run under the gathers below.
    if (tid < 32) {
        u64 ga = (u64)(uintptr_t)poses + (u64)base * 28u;   // tile start (bytes)
        u32x4 g0;
        g0[0] = 1u;                                          // count=1, user mode
        g0[1] = 0u;                                          // lds_addr = 0 (sposes base)
        g0[2] = (u32)(ga & 0xFFFFFFFFull);                   // global_addr[31:0]
        g0[3] = (u32)((ga >> 32) & 0x01FFFFFFull) | (2u << 30); // addr[56:32] | type=2
        i32x8 g1;
        g1[0] = (int)(2u << 16);                             // data_size=4B, mask=0, no flags
        g1[1] = (int)((total_pose_dw & 0xFFFFu) << 16);      // tensor_dim0[15:0]
        g1[2] = (int)((total_pose_dw >> 16) & 0xFFFFu);      // tensor_dim0[31:16], dim1=0
        g1[3] = (int)((u32)POSE_DW << 16);                   // tile_dim0 = 1792 (1-D tile)
        g1[4] = 0;                                           // tile_dim1/2 unused
        g1[5] = (int)total_pose_dw;                          // dim0 stride (benign for 1-D)
        g1[6] = 0;
        g1[7] = 0;
        i32x4 gz4 = {0, 0, 0, 0};                            // groups 2/3 unused (<=2D)
        i32x8 gz8 = {0, 0, 0, 0, 0, 0, 0, 0};
        __builtin_amdgcn_tensor_load_to_lds(g0, g1, gz4, gz4, gz8, 0);
    }

    // ---- Stage 2: issue all LDS-independent loads NOW, overlapped with the DMA.
    //      Index is clamped (not early-returned) so every thread reaches the barrier.
    const int i  = base + tid;
    const bool live = (i < n_edges);
    const int iv = live ? i : (n_edges - 1);

    // Edge indices: streaming, non-temporal (don't pollute L2 holding the node table)
    long long e = __builtin_nontemporal_load(edges_pk + iv);
    int i0 = (int)(e & 0xFFFFFFFFll);
    int i1 = (int)((u64)e >> 32);

    // Node gathers: small hot table (5.6 MB), regular-temporal loads (L2/WGP$ resident)
    const float* pa = nodes + (size_t)i0 * 7;
    const float* pb = nodes + (size_t)i1 * 7;
    f3   ta{pa[0], pa[1], pa[2]};
    quat qa{pa[3], pa[4], pa[5], pa[6]};
    f3   tb{pb[0], pb[1], pb[2]};
    quat qb{pb[3], pb[4], pb[5], pb[6]};

    // ---- Stage 3: drain the DMA (no-op for non-issuing waves) and sync the block.
    __builtin_amdgcn_s_wait_tensorcnt(0);
    __syncthreads();

    // Pose for this edge from LDS (stride-7 dwords: conflict-free across 64 banks)
    const float* sp = sposes + tid * 7;
    f3   tp{sp[0], sp[1], sp[2]};
    quat qp{sp[3], sp[4], sp[5], sp[6]};

    // inv(node1): qi = conj(qa); ti = -qrot(qi, ta)
    quat qai{-qa.x, -qa.y, -qa.z, qa.w};
    f3 tr = qrot(qai, ta);
    f3 tai{-tr.x, -tr.y, -tr.z};

    // T1 = pose ∘ inv(node1)
    f3 rt = qrot(qp, tai);
    f3 t1{tp.x + rt.x, tp.y + rt.y, tp.z + rt.z};
    quat q1 = qmul(qp, qai);

    // T = T1 ∘ node2
    f3 rb = qrot(q1, tb);
    f3 t{t1.x + rb.x, t1.y + rb.y, t1.z + rb.z};
    quat q = qmul(q1, qb);

    // ---- SE(3) log map ----
    float s = (q.w < 0.f) ? -1.f : 1.f;          // canonicalize w >= 0
    q.x *= s; q.y *= s; q.z *= s; q.w *= s;
    float nn2   = q.x*q.x + q.y*q.y + q.z*q.z;
    float n     = sqrtf(nn2);
    float theta = 2.f * atan2f(n, q.w);
    float scale = (n < 1e-6f) ? (2.f / fmaxf(q.w, 1e-12f)) : (theta / n);
    f3 phi{scale*q.x, scale*q.y, scale*q.z};
    float th2 = theta * theta;
    // Unit-quaternion identity: (1+cosθ)/(2θ sinθ) = w/(2θn)  -> no sinf/cosf needed
    float coef = (theta < 1e-4f) ? (1.f/12.f + th2 * (1.f/720.f))
                                 : (1.f/th2 - q.w / (2.f * theta * n));
    f3 pxt  = cross3(phi, t);
    f3 ppxt = cross3(phi, pxt);
    f3 rho{ t.x - 0.5f*pxt.x + coef*ppxt.x,
            t.y - 0.5f*pxt.y + coef*ppxt.y,
            t.z - 0.5f*pxt.z + coef*ppxt.z };

    // Output: pure 96 MB stream -> non-temporal stores (guarded for the tail block)
    if (live) {
        float* o = out + (size_t)i * 6;
        __builtin_nontemporal_store(rho.x, o + 0);
        __builtin_nontemporal_store(rho.y, o + 1);
        __builtin_nontemporal_store(rho.z, o + 2);
        __builtin_nontemporal_store(phi.x, o + 3);
        __builtin_nontemporal_store(phi.y, o + 4);
        __builtin_nontemporal_store(phi.z, o + 5);
    }
}

extern "C" void kernel_launch(void* const* d_in, const int* in_sizes, int n_in,
                              void* d_out, int out_size, void* d_ws, size_t ws_size,
                              hipStream_t stream) {
    const float*     nodes = (const float*)d_in[0];
    const float*     poses = (const float*)d_in[1];
    const long long* edges = (const long long*)d_in[2];  // int32 pairs, read packed
    float* out = (float*)d_out;

    const int n_edges       = in_sizes[2] / 2;           // (E, 2) index array
    const u32 total_pose_dw = (u32)in_sizes[1];          // E * 7 dwords
    const int blocks        = (n_edges + BLOCK - 1) / BLOCK;
    if (blocks <= 0) return;

    posegraph_err_kernel<<<blocks, BLOCK, 0, stream>>>(nodes, poses, edges, out,
                                                       n_edges, total_pose_dw);
}